// BLT_721554506095
// MI455X (gfx1250) — compile-verified
//
#include <hip/hip_runtime.h>

// ---------------------------------------------------------------------------
// MI455X (gfx1250, wave32) implementation of the entropy-NMS patch-reassembly
// pipeline.  Memory-roofline dominated (~470 MB traffic @ 23.3 TB/s ≈ 20 us);
// the 16^3 ones-conv is expressed as banded-matrix matmuls via
// V_WMMA_F32_16X16X4_F32 (exact f32 math).  NMS uses a hierarchical argmax
// (per-32-word maxima + wave32 shuffle butterflies + ballot) so the serial
// 216-round loop needs only 4 barriers and ~3K LDS reads per round.
// ---------------------------------------------------------------------------

typedef float v2f __attribute__((ext_vector_type(2)));
typedef float v8f __attribute__((ext_vector_type(8)));

#define IMG   96
#define VOX   884736      // 96^3
#define PATCH 16
#define GRID  41
#define M_WIN 68921       // 41^3
#define NSEL  216

// ---------------------------------------------------------------------------
// Kernel 1: per-voxel entropy over 16 channels.
// E[b][v] = -sum_c p*log2(p + 1e-10)      (p read once -> non-temporal)
// ---------------------------------------------------------------------------
__global__ void entropy_kernel(const float* __restrict__ p, float* __restrict__ E) {
    long i = (long)blockIdx.x * 256 + threadIdx.x;   // over 2*VOX
    if (i >= 2L * VOX) return;
    long b = i / VOX, v = i - b * VOX;
    const float* pp = p + b * 16L * VOX + v;
    float s = 0.0f;
#pragma unroll
    for (int ch = 0; ch < 16; ++ch) {
        float q = __builtin_nontemporal_load(pp + (long)ch * VOX);
        s -= q * log2f(q + 1e-10f);
    }
    E[i] = s;
}

// ---------------------------------------------------------------------------
// Kernel 2/3/4: generic 1-D band-sum pass (window 16, stride 2) as WMMA.
//   out[m][n] = sum_{k<16} in[2m+k][n],  m in [0,41), reduce-dim length 96.
// Per 16x16 tile:  D = A(16x48 0/1 band) x B(48x16 data) as 12 chained
// V_WMMA_F32_16X16X4_F32 (K=4 each), f32-exact.
// "in" element (r,n) at plane + r*sR + n*sN; out (m,n) at m*sMout + n*sNout.
// ---------------------------------------------------------------------------
__global__ void band_conv_wmma(const float* __restrict__ in, float* __restrict__ out,
                               int nColTiles, int nPlanes,
                               long planeInStride, long planeOutStride,
                               int sR, int sN, int C,
                               int sMout, int sNout)
{
    int wave = (int)((blockIdx.x * blockDim.x + threadIdx.x) >> 5);
    int lane = threadIdx.x & 31;
    int tilesPerPlane = 3 * nColTiles;           // 3 m-tiles cover 41 rows
    int total = nPlanes * tilesPerPlane;
    if (wave >= total) return;                   // wave-uniform: EXEC stays all-1s

    int plane = wave / tilesPerPlane;
    int rem   = wave - plane * tilesPerPlane;
    int mTile = rem / nColTiles;
    int nTile = rem - mTile * nColTiles;
    int m0 = mTile * 16;
    int n0 = nTile * 16;

    int j  = lane & 15;     // A-row (M) index AND B-column (N) index for this lane
    int hi = lane >> 4;     // lane half selects K pair {0,1} vs {2,3} per ISA layout

    const float* pin  = in  + (long)plane * planeInStride;
    float*       pout = out + (long)plane * planeOutStride;

    int  nn   = n0 + j;
    bool ncl  = nn < C;
    long colB = (long)nn * sN;

    v8f acc = {0.f, 0.f, 0.f, 0.f, 0.f, 0.f, 0.f, 0.f};

#pragma unroll
    for (int c = 0; c < 12; ++c) {
        int r0 = 4 * c + 2 * hi;                 // this lane's two K indices: r0, r0+1
        // A[m=j][r] = 1 iff 2j <= r < 2j+16   (the stride-2 window band)
        v2f a;
        a.x = ((unsigned)(r0     - 2 * j) < 16u) ? 1.0f : 0.0f;
        a.y = ((unsigned)(r0 + 1 - 2 * j) < 16u) ? 1.0f : 0.0f;
        // B[r][n=j] = in[2*m0 + r][nn], clamped to reduce-dim < 96
        int rr = 2 * m0 + r0;
        v2f bv;
        bv.x = (ncl && rr     < IMG) ? pin[colB + (long)(rr    ) * sR] : 0.0f;
        bv.y = (ncl && rr + 1 < IMG) ? pin[colB + (long)(rr + 1) * sR] : 0.0f;
        acc = __builtin_amdgcn_wmma_f32_16x16x4_f32(
            /*neg_a=*/false, a, /*neg_b=*/false, bv,
            /*c_mod=*/(short)0, acc, /*reuse_a=*/false, /*reuse_b=*/false);
    }

    // D layout: lane holds D[M = v + 8*hi][N = j]
    if (ncl) {
#pragma unroll
        for (int v = 0; v < 8; ++v) {
            int m = m0 + v + 8 * hi;
            if (m < GRID)
                pout[(long)m * sMout + (long)nn * sNout] = acc[v];
        }
    }
}

// ---------------------------------------------------------------------------
// Kernel 5: greedy NMS, one 256-thread workgroup per batch.
// Hierarchical argmax: LDS per-word maxima (2154 words of 32 scores),
// wave32 shuffle butterfly reductions, ballot for the in-word element.
// Exact reference semantics: max value -> min word -> min bit (first index).
// Suppress iff integer overlap volume ov > 1365  (<=> ov/(8192-ov) > 0.2).
// 4 __syncthreads per round.
// ---------------------------------------------------------------------------
#define NMS_WORDS 2154   // ceil(68921/32)
#define NMS_T     256

__global__ void __launch_bounds__(NMS_T)
nms_kernel(const float* __restrict__ S3, int* __restrict__ sel) {
    int b = blockIdx.x;
    const float* sc = S3 + (long)b * M_WIN;
    __shared__ unsigned int  validw[NMS_WORDS];
    __shared__ float         wmax[NMS_WORDS];
    __shared__ unsigned char aff[NMS_WORDS];
    __shared__ float redv[8];
    __shared__ int   redw[8];
    __shared__ int   sWidx;

    int t    = threadIdx.x;
    int lane = t & 31;
    int wv   = t >> 5;
    const float NEG = -__builtin_huge_valf();

    for (int w = t; w < NMS_WORDS; w += NMS_T) {
        validw[w] = (w == NMS_WORDS - 1) ? ((1u << (M_WIN - 32 * (NMS_WORDS - 1))) - 1u)
                                         : 0xFFFFFFFFu;
        aff[w] = 0;
    }
    __syncthreads();
    // initial word maxima (single full pass over scores)
    for (int w = t; w < NMS_WORDS; w += NMS_T) {
        unsigned int vm = validw[w];
        float mv = NEG;
        int base = w * 32;
        while (vm) {
            int q = __builtin_ctz(vm);
            vm &= vm - 1;
            float v = sc[base + q];
            if (v > mv) mv = v;
        }
        wmax[w] = mv;
    }
    __syncthreads();

    for (int it = 0; it < NSEL; ++it) {
        // ---- phase 1: scan word maxima; strict > keeps earliest word per thread
        float bv = NEG;
        int   bw = 0x7FFFFFFF;
        for (int w = t; w < NMS_WORDS; w += NMS_T) {
            float v = wmax[w];
            if (v > bv) { bv = v; bw = w; }
        }
        // wave32 butterfly (tie -> min word)
        for (int m = 1; m < 32; m <<= 1) {
            float ov = __shfl_xor(bv, m, 32);
            int   ow = __shfl_xor(bw, m, 32);
            if (ov > bv || (ov == bv && ow < bw)) { bv = ov; bw = ow; }
        }
        if (lane == 0) { redv[wv] = bv; redw[wv] = bw; }
        __syncthreads();                                   // barrier 1

        // ---- phase 2: wave 0 combines 8 wave results, then ballot in-word
        if (wv == 0) {
            float v8 = (lane < 8) ? redv[lane] : NEG;
            int   w8 = (lane < 8) ? redw[lane] : 0x7FFFFFFF;
            for (int m = 1; m < 8; m <<= 1) {
                float ov = __shfl_xor(v8, m, 32);
                int   ow = __shfl_xor(w8, m, 32);
                if (ov > v8 || (ov == v8 && ow < w8)) { v8 = ov; w8 = ow; }
            }
            v8 = __shfl(v8, 0, 32);
            w8 = __shfl(w8, 0, 32);
            // first element of the winning word achieving its max
            int gi = 32 * w8 + lane;
            bool hit = (gi < M_WIN) && ((validw[w8] >> lane) & 1u) && (sc[gi] == v8);
            unsigned long long bal = __ballot(hit);
            int q = bal ? (__ffsll((unsigned long long)bal) - 1) : 0;
            if (lane == 0) {
                int widx = 32 * w8 + q;
                sWidx = widx;
                sel[b * NSEL + it] = widx;
            }
        }
        __syncthreads();                                   // barrier 2

        int idx = sWidx;
        int xs = idx / 1681;
        int rm = idx - xs * 1681;
        int ys = rm / GRID;
        int zs = rm - ys * GRID;

        // ---- phase 3: suppression.  Per (x2,y2) row the suppressed z-range is
        // contiguous: need ox*oy*oz > 1365 with o = 16-2|d|, |d|<=5.
        int xlo = max(xs - 5, 0), xhi = min(xs + 5, GRID - 1);
        int ylo = max(ys - 5, 0), yhi = min(ys + 5, GRID - 1);
        int ny = yhi - ylo + 1;
        int nrows = (xhi - xlo + 1) * ny;
        for (int r = t; r < nrows; r += NMS_T) {
            int x2 = xlo + r / ny;
            int y2 = ylo + r % ny;
            int ox = 16 - 2 * abs(x2 - xs);
            int oy = 16 - 2 * abs(y2 - ys);
            int oxy = ox * oy;
            if (oxy * 16 <= 1365) continue;     // even oz=16 can't suppress
            int ozmin = 1365 / oxy + 1;         // smallest suppressing oz
            int dz = (16 - ozmin) >> 1;         // |dz| <= dz  => contiguous z band
            int zlo = max(zs - dz, 0), zhi = min(zs + dz, GRID - 1);
            int g0 = x2 * 1681 + y2 * GRID + zlo;
            int g1 = g0 + (zhi - zlo);
            int w0 = g0 >> 5, w1 = g1 >> 5;
            for (int w = w0; w <= w1; ++w) {
                int lo  = max(g0 - w * 32, 0);
                int hi2 = min(g1 - w * 32, 31);
                unsigned int mask = (hi2 == 31 ? 0xFFFFFFFFu : ((1u << (hi2 + 1)) - 1u))
                                    & ~((1u << lo) - 1u);
                atomicAnd(&validw[w], ~mask);
                aff[w] = 1;
            }
        }
        __syncthreads();                                   // barrier 3

        // ---- phase 4: recompute maxima only for touched words
        int wA = (xlo * 1681) >> 5;
        int wB = (xhi * 1681 + 1680) >> 5;
        for (int w = wA + t; w <= wB; w += NMS_T) {
            if (!aff[w]) continue;
            aff[w] = 0;
            unsigned int vm = validw[w];
            float mv = NEG;
            int base = w * 32;
            while (vm) {
                int q2 = __builtin_ctz(vm);
                vm &= vm - 1;
                float v = sc[base + q2];
                if (v > mv) mv = v;
            }
            wmax[w] = mv;
        }
        __syncthreads();                                   // barrier 4
    }
}

// ---------------------------------------------------------------------------
// Kernel 6: patch gather + reassembly + residual add.  The 216 selected 16^3
// patches tile the 96^3 output exactly once, so one pass writes every voxel:
//   out[b,c, 16h+p1, 16w+p2, 16d+p3] = x[b,c,dst] + x[b,c, s0+p1, s1+p2, s2+p3]
// One block per (b, n, ch); float2 accesses (starts even -> 8B aligned).
// Dest-x reads and out writes are stream-once -> non-temporal hints.
// ---------------------------------------------------------------------------
__global__ void gather_kernel(const float* __restrict__ x,
                              const int* __restrict__ sel,
                              float* __restrict__ out)
{
    int blk = blockIdx.x;                 // = ((b*216)+n)*16 + ch
    int ch  = blk & 15;
    int bn  = blk >> 4;
    int n   = bn % NSEL;
    int b   = bn / NSEL;

    int idx = sel[b * NSEL + n];
    int xs  = idx / 1681;
    int rm  = idx - xs * 1681;
    int ys  = rm / GRID;
    int zs  = rm - ys * GRID;
    int s0 = 2 * xs, s1 = 2 * ys, s2 = 2 * zs;

    int h = n / 36, w = (n / 6) % 6, d = n % 6;

    const float* xb = x   + ((long)(b * 16 + ch)) * VOX;
    float*       ob = out + ((long)(b * 16 + ch)) * VOX;

#pragma unroll
    for (int itr = 0; itr < 8; ++itr) {
        int f  = itr * 256 + threadIdx.x;       // float2 index in [0,2048)
        int p1 = f >> 7;
        int p2 = (f >> 3) & 15;
        int q  = f & 7;
        long src = ((long)(s0 + p1) * IMG + (s1 + p2)) * IMG + s2 + 2 * q;
        long dst = ((long)(16 * h + p1) * IMG + (16 * w + p2)) * IMG + 16 * d + 2 * q;
        v2f sv = *(const v2f*)(xb + src);                       // may be reused (overlaps)
        v2f dv = __builtin_nontemporal_load((const v2f*)(xb + dst));
        v2f ov = sv + dv;
        __builtin_nontemporal_store(ov, (v2f*)(ob + dst));
    }
}

// ---------------------------------------------------------------------------
// Launch.  Workspace layout (floats):
//   E  [2*884736]  S1 [2*96*96*41]  S2 [2*96*41*41]  S3 [2*41^3]  sel [432 int]
// Total ~12 MB.
// ---------------------------------------------------------------------------
extern "C" void kernel_launch(void* const* d_in, const int* in_sizes, int n_in,
                              void* d_out, int out_size, void* d_ws, size_t ws_size,
                              hipStream_t stream) {
    const float* x = (const float*)d_in[0];
    const float* p = (const float*)d_in[1];
    // d_in[2] (weights) is all-ones in the reference; the band matrix encodes it.

    float* ws = (float*)d_ws;
    float* E   = ws;                       // 1,769,472
    float* S1  = E  + 1769472;             //   755,712  (b,96,96,41)
    float* S2  = S1 + 755712;              //   322,752  (b,96,41,41)
    float* S3  = S2 + 322752;              //   137,842  (b,41,41,41)
    int*   sel = (int*)(S3 + 137842);      //       432
    float* out = (float*)d_out;

    // 1) entropy: 2*96^3 voxels
    entropy_kernel<<<6912, 256, 0, stream>>>(p, E);

    // 2) z-pass: 1 plane; reduce stride 1; 18432 columns (b*x*y rows) stride 96.
    //    out S1[row][zs]: m=zs stride 1, n=row stride 41.   3456 waves.
    band_conv_wmma<<<432, 256, 0, stream>>>(E, S1, 1152, 1, 0L, 0L,
                                            1, 96, 18432, 1, 41);
    // 3) y-pass: 192 (b,x) planes; reduce stride 41; 41 zs columns stride 1.
    //    out S2[ys][zs].   1728 waves.
    band_conv_wmma<<<216, 256, 0, stream>>>(S1, S2, 3, 192, 3936L, 1681L,
                                            41, 1, 41, 41, 1);
    // 4) x-pass: 2 b planes; reduce stride 1681; 1681 (ys,zs) columns stride 1.
    //    out S3[xs][ys*41+zs].   636 waves.
    band_conv_wmma<<<80, 256, 0, stream>>>(S2, S3, 106, 2, 161376L, 68921L,
                                           1681, 1, 1681, 1681, 1);

    // 5) greedy NMS, one workgroup per batch (hierarchical argmax)
    nms_kernel<<<2, NMS_T, 0, stream>>>(S3, sel);

    // 6) gather + reassemble + residual add (writes every output voxel once)
    gather_kernel<<<6912, 256, 0, stream>>>(x, sel, out);
}